// DeepTermRankingListNet_52501680226474
// MI455X (gfx1250) — compile-verified
//
#include <hip/hip_runtime.h>
#include <hip/hip_bf16.h>
#include <math.h>

// ---------------------------------------------------------------------------
// DeepTermRankingListNet on MI455X (gfx1250)
//   C=256 candidates, K=64 contexts, d=128, Ds=200
//   Main compute: per-candidate 64x64x128 f32 GEMM -> tanh -> softmax ->
//   weighted embeddings -> bilinear score; blended with string cosine score.
//   GEMM runs on V_WMMA_F32_16X16X4_F32 (full f32, matches reference RNE).
// ---------------------------------------------------------------------------

typedef __attribute__((ext_vector_type(2))) float v2f;
typedef __attribute__((ext_vector_type(8))) float v8f;

#define DTR_V 500000
#define DTR_D 128
#define DTR_K 64
#define DTR_C 256
#define DTR_DS 200
#define DTR_EPS 1e-8f
#define DTR_GAMMA 0.5f

// ---------------------------------------------------------------------------
// Kernel 1: gather A = table[t1_ctx] (-> ws) and compute AM = A @ att_mat (-> ws)
//   grid = K (64) blocks, block = d (128) threads; block k handles row k.
// ---------------------------------------------------------------------------
__global__ __launch_bounds__(DTR_D) void dtr_prep_kernel(
    const float* __restrict__ table,
    const float* __restrict__ att_mat,
    const int* __restrict__ t1_ctx,
    float* __restrict__ AM,      // [K, d]
    float* __restrict__ Aout)    // [K, d]
{
    __shared__ float sA[DTR_D];
    const int k = blockIdx.x;    // context row 0..63
    const int j = threadIdx.x;   // output column 0..127

    const long long row = (long long)t1_ctx[k] * DTR_D;
    const float a = table[row + j];
    sA[j] = a;
    Aout[k * DTR_D + j] = a;
    __syncthreads();

    float acc = 0.0f;
    #pragma unroll 8
    for (int dd = 0; dd < DTR_D; ++dd) {
        acc = fmaf(sA[dd], att_mat[dd * DTR_D + j], acc);
    }
    AM[k * DTR_D + j] = acc;
}

// ---------------------------------------------------------------------------
// Kernel 2: one workgroup per candidate c.
//   512 threads = 16 waves. Wave w computes sim tile (tm,tn) = 16*(w/4, w%4)
//   with V_WMMA_F32_16X16X4_F32 accumulating over K=128 in steps of 4.
// ---------------------------------------------------------------------------
__global__ __launch_bounds__(512) void dtr_main_kernel(
    const float* __restrict__ table,
    const float* __restrict__ str_t1,   // [Ds]
    const float* __restrict__ str_t2s,  // [C, Ds]
    const float* __restrict__ W_bi,     // [d, d]
    const float* __restrict__ b_bi,     // [1]
    const int* __restrict__ t2_ctx,     // [C, K]
    const float* __restrict__ AM,       // [K, d] (ws)
    const float* __restrict__ A,        // [K, d] (ws)
    float* __restrict__ out)            // [C]
{
    __shared__ float sB[DTR_K][DTR_D];     // gathered B[c]   : 32 KB
    __shared__ float sSim[DTR_K][DTR_K];   // tanh(AM.B^T)    : 16 KB
    __shared__ float sRows[DTR_K];         // softmax over k
    __shared__ float sCols[DTR_K];         // softmax over m
    __shared__ float sNewA[DTR_D];
    __shared__ float sNewB[DTR_D];
    __shared__ float sRed[512];            // block reduction : 2 KB
    __shared__ float sDot[256], sN2p[256], sN1p[256];  // string branch : 3 KB
    __shared__ float sStrScore;

    const int c    = blockIdx.x;
    const int tid  = threadIdx.x;
    const int lane = tid & 31;
    const int wave = tid >> 5;

    // ---- string branch partials (cosine similarity over Ds=200) ----
    {
        float pd = 0.0f, p1 = 0.0f, p2 = 0.0f;
        if (tid < DTR_DS) {
            const float x1 = str_t1[tid];
            const float x2 = str_t2s[c * DTR_DS + tid];
            pd = x1 * x2;
            p1 = x1 * x1;
            p2 = x2 * x2;
        }
        if (tid < 256) { sDot[tid] = pd; sN1p[tid] = p1; sN2p[tid] = p2; }
    }

    // ---- gather B[c] rows into LDS: 64 rows x 128 f32, 16 floats/thread ----
    {
        const int row = tid >> 3;   // 0..63
        const int seg = tid & 7;    // 0..7 (16 floats each)
        const long long src = (long long)t2_ctx[c * DTR_K + row] * DTR_D;
        const float4* sp = (const float4*)(table + src + seg * 16);
        float4* dp = (float4*)(&sB[row][seg * 16]);
        dp[0] = sp[0]; dp[1] = sp[1]; dp[2] = sp[2]; dp[3] = sp[3];
    }
    __syncthreads();

    // thread 0 finalizes string score while others start the GEMM region
    if (tid == 0) {
        float dsum = 0.0f, n1s = 0.0f, n2s = 0.0f;
        for (int i = 0; i < 256; ++i) { dsum += sDot[i]; n1s += sN1p[i]; n2s += sN2p[i]; }
        const float n1 = fmaxf(sqrtf(n1s), DTR_EPS);
        const float n2 = fmaxf(sqrtf(n2s), DTR_EPS);
        sStrScore = dsum / (n1 * n2);
    }

    // ---- sim = tanh(AM @ B[c]^T) via f32 WMMA, 16x16 tile per wave ----
    {
        const int tm   = (wave >> 2) << 4;   // 0,16,32,48
        const int tn   = (wave & 3) << 4;    // 0,16,32,48
        const int half = lane >> 4;          // K-pair selector
        const int l16  = lane & 15;          // M (for A) / N (for B)

        const float* amRow = AM + (long long)(tm + l16) * DTR_D; // A operand: M = tm+l16
        const float* bRow  = &sB[tn + l16][0];                   // B operand: N = tn+l16

        v8f acc = {};
        #pragma unroll 4
        for (int kk = 0; kk < DTR_D; kk += 4) {
            v2f av, bv;
            av.x = amRow[kk + 2 * half + 0];
            av.y = amRow[kk + 2 * half + 1];
            bv.x = bRow[kk + 2 * half + 0];
            bv.y = bRow[kk + 2 * half + 1];
            acc = __builtin_amdgcn_wmma_f32_16x16x4_f32(
                /*neg_a=*/false, av, /*neg_b=*/false, bv,
                /*c_mod=*/(short)0, acc, /*reuse_a=*/false, /*reuse_b=*/false);
        }
        // D layout: VGPR r, lane -> M = tm + r + 8*half, N = tn + l16
        #pragma unroll
        for (int r = 0; r < 8; ++r) {
            sSim[tm + r + 8 * half][tn + l16] = tanhf(acc[r]);
        }
    }
    __syncthreads();

    // ---- row/col means over the 64x64 sim tile ----
    if (tid < DTR_K) {
        float rs = 0.0f, cs = 0.0f;
        for (int m = 0; m < DTR_K; ++m) rs += sSim[tid][m];  // mean over m (axis=2)
        for (int k = 0; k < DTR_K; ++k) cs += sSim[k][tid];  // mean over k (axis=1)
        sRows[tid] = rs * (1.0f / DTR_K);
        sCols[tid] = cs * (1.0f / DTR_K);
    }
    __syncthreads();

    // ---- softmax (serial, 64 values: negligible) ----
    if (tid == 0) {
        float mx = sRows[0];
        for (int k = 1; k < DTR_K; ++k) mx = fmaxf(mx, sRows[k]);
        float den = 0.0f;
        for (int k = 0; k < DTR_K; ++k) { const float e = expf(sRows[k] - mx); sRows[k] = e; den += e; }
        const float inv = 1.0f / den;
        for (int k = 0; k < DTR_K; ++k) sRows[k] *= inv;
    } else if (tid == 32) {
        float mx = sCols[0];
        for (int k = 1; k < DTR_K; ++k) mx = fmaxf(mx, sCols[k]);
        float den = 0.0f;
        for (int k = 0; k < DTR_K; ++k) { const float e = expf(sCols[k] - mx); sCols[k] = e; den += e; }
        const float inv = 1.0f / den;
        for (int k = 0; k < DTR_K; ++k) sCols[k] *= inv;
    }
    __syncthreads();

    // ---- new_A = rows @ A, new_B = cols @ B[c] ----
    if (tid < DTR_D) {
        float sa = 0.0f, sb = 0.0f;
        for (int k = 0; k < DTR_K; ++k) {
            sa = fmaf(sRows[k], A[k * DTR_D + tid], sa);
            sb = fmaf(sCols[k], sB[k][tid], sb);
        }
        sNewA[tid] = sa;
        sNewB[tid] = sb;
    }
    __syncthreads();

    // ---- bilinear: sum_e (sum_d newA[d] * W[d][e]) * newB[e] ----
    float part = 0.0f;
    if (tid < DTR_D) {
        float t = 0.0f;
        #pragma unroll 8
        for (int dd = 0; dd < DTR_D; ++dd) {
            t = fmaf(sNewA[dd], W_bi[dd * DTR_D + tid], t);
        }
        part = t * sNewB[tid];
    }
    sRed[tid] = part;
    __syncthreads();
    for (int s = 256; s > 0; s >>= 1) {
        if (tid < s) sRed[tid] += sRed[tid + s];
        __syncthreads();
    }

    if (tid == 0) {
        const float con = sRed[0] + b_bi[0];
        out[c] = (1.0f - DTR_GAMMA) * sStrScore + DTR_GAMMA * con;
    }
}

// ---------------------------------------------------------------------------
// Launch
// ---------------------------------------------------------------------------
extern "C" void kernel_launch(void* const* d_in, const int* in_sizes, int n_in,
                              void* d_out, int out_size, void* d_ws, size_t ws_size,
                              hipStream_t stream) {
    const float* table   = (const float*)d_in[0];
    const float* str_t1  = (const float*)d_in[1];
    const float* str_t2s = (const float*)d_in[2];
    const float* att_mat = (const float*)d_in[3];
    const float* W_bi    = (const float*)d_in[4];  // [1,d,d] -> [d,d]
    const float* b_bi    = (const float*)d_in[5];
    const int*   t1_ctx  = (const int*)d_in[6];
    const int*   t2_ctx  = (const int*)d_in[7];
    float*       out     = (float*)d_out;

    float* AM = (float*)d_ws;            // [K, d]
    float* A  = AM + DTR_K * DTR_D;      // [K, d]

    dtr_prep_kernel<<<DTR_K, DTR_D, 0, stream>>>(table, att_mat, t1_ctx, AM, A);
    dtr_main_kernel<<<DTR_C, 512, 0, stream>>>(table, str_t1, str_t2s, W_bi, b_bi,
                                               t2_ctx, AM, A, out);
}